// ThermalMassLoss_28802050687858
// MI455X (gfx1250) — compile-verified
//
#include <hip/hip_runtime.h>
#include <hip/hip_bf16.h>
#include <cstdint>

#define DT_STEP 0.04f
#define CHUNK   4096
#define BLOCK   256
#define PT      (CHUNK / BLOCK)          // 16 elements per thread
#define LDSN    (1 + CHUNK + CHUNK / 16) // halo + padded tile (1 dword pad / 16 dwords)

typedef unsigned int v4u   __attribute__((ext_vector_type(4)));
typedef int          v8i_t __attribute__((ext_vector_type(8)));
typedef int          v4i_t __attribute__((ext_vector_type(4)));

// gsw_sp_from_c depends only on conductivity C (T,P unused in the reference).
__device__ __forceinline__ float sp_from_c(float c) {
    float rt = __builtin_sqrtf(c * (1.0f / 42.914f));
    float s = ((((0.27081f * rt - 0.70261f) * rt + 0.14692f) * rt
                + (0.025986f - 0.0060269f * rt) * rt) * rt + 0.35f) * rt;
    return s * 35.0f;
}

// Padded LDS index for tile-local element i: TDM inserts 1 dword of pad every
// 16 dwords (pad_interval=3, pad_amount=0), so element i sits at i + i/16.
// Lane t reads 17t+e mod 64 -> bank-conflict-free (gcd(17,64)=1).
__device__ __forceinline__ int padidx(int i) { return i + (i >> 4); }

// TDM 1-D tile load: nelem fp32 from global -> LDS with bank-deconflict padding.
// D# per CDNA5 ISA: group0 {count=1, lds_addr, 57-bit global addr, type=2};
// group1 {data_size=4B, pad_enable, pad_interval=16dw, pad_amount=1dw,
// tensor_dim0 = tile_dim0 = nelem, dim1 = 1}. Per-wave issue, EXEC-ignored.
__device__ __forceinline__ void tdm_load_1d(unsigned lds_byte_addr,
                                            const float* gptr, unsigned nelem) {
    unsigned long long ga = (unsigned long long)(uintptr_t)gptr;
    v4u g0;
    g0[0] = 1u;                                               // count=1, user mode
    g0[1] = lds_byte_addr;                                    // lds_addr [63:32]
    g0[2] = (unsigned)ga;                                     // global_addr lo
    g0[3] = (unsigned)((ga >> 32) & 0x01FFFFFFull) | (2u << 30); // addr hi | type=2
    v8i_t g1;
    g1[0] = (int)(0x00020000u      // data_size=2 (4B)
                | (1u << 20)       // pad_enable
                | (3u << 22));     // pad_interval=3 -> every 16 DWORDs (+1 DWORD pad)
    g1[1] = (int)((nelem & 0xFFFFu) << 16);                   // tensor_dim0 lo16 @ [63:48]
    g1[2] = (int)(((nelem >> 16) & 0xFFFFu) | (1u << 16));    // tensor_dim0 hi | tensor_dim1=1
    g1[3] = (int)((nelem & 0xFFFFu) << 16);                   // tile_dim0 @ [127:112]
    g1[4] = 1;                                                // tile_dim1=1, tile_dim2=0
    g1[5] = (int)nelem;                                       // tensor_dim0_stride lo32
    g1[6] = 0;
    g1[7] = 0;
    v4i_t z4 = {0, 0, 0, 0};
    v8i_t z8 = {0, 0, 0, 0, 0, 0, 0, 0};
    __builtin_amdgcn_tensor_load_to_lds(g0, g1, z4, z4, z8, 0);
}

// Chunked affine-scan kernel. FINAL=false: emit per-chunk (A,B) aggregate.
// FINAL=true: apply carry-in, produce y, fold masked-MSE partial sums.
template <bool FINAL>
__global__ __launch_bounds__(BLOCK)
void garau_scan_kernel(const float* __restrict__ T, const float* __restrict__ C,
                       const float* __restrict__ Sref,
                       const unsigned char* __restrict__ mask,
                       const float* __restrict__ params, int poff,
                       int seqLen, int nchunks,
                       float* __restrict__ agg,
                       const float* __restrict__ carry,
                       float* __restrict__ partial) {
    __shared__ float ldsT[LDSN];   // [0] = halo T[start-1], tile at [1..] padded
    __shared__ float ldsC[LDSN];   // [0] = halo C[start-1], tile at [1..] padded
    __shared__ float sA[BLOCK];
    __shared__ float sB[BLOCK];

    const int tid     = threadIdx.x;
    const int row     = blockIdx.x / nchunks;
    const int chk     = blockIdx.x % nchunks;
    const int start   = chk * CHUNK;
    const int rowbase = row * seqLen;

    const float alpha0 = params[row * 8 + poff + 0];
    const float alphaS = params[row * 8 + poff + 1];
    const float tau0   = params[row * 8 + poff + 2];
    const float tauS   = params[row * 8 + poff + 3];

    if (tid == 0) {
        int h = (start > 0) ? (start - 1) : 0;   // halo (unused for global elem 0)
        ldsT[0] = T[rowbase + h];
        ldsC[0] = C[rowbase + h];
    }
    if (tid < 32) {  // wave 0 only: one TDM issue per descriptor
        tdm_load_1d((unsigned)(uintptr_t)&ldsT[1], T + rowbase + start, CHUNK);
        tdm_load_1d((unsigned)(uintptr_t)&ldsC[1], C + rowbase + start, CHUNK);
        __builtin_amdgcn_s_wait_tensorcnt(0);
    }
    __syncthreads();

    // Per-element affine coefficients y[i] = Ae*y[i-1] + Be  (i chunk-local)
    auto elemAB = [&](int i, float& Ae, float& Be) {
        const int ic = 1 + padidx(i);                       // C[i], T[i]
        const int ip = (i == 0) ? 0 : 1 + padidx(i - 1);    // C[i-1], T[i-1] (or halo)
        float Ci = ldsC[ic];
        float Sx = sp_from_c(Ci);
        float alpha = alpha0 + alphaS * Sx;
        float tau   = tau0 + tauS * Sx;
        float a     = __expf(-DT_STEP / tau);
        if (start + i == 0) {           // first element of the row
            Ae = 0.0f; Be = Ci;
        } else {
            float dTdt = (ldsT[ic] - ldsT[ip]) * (1.0f / DT_STEP);
            Ae = -a;
            Be = Ci + alpha * tau * dTdt + a * ldsC[ip];
        }
    };

    // Thread-serial composition over PT contiguous elements
    const int t0 = tid * PT;
    float Ap = 1.0f, Bp = 0.0f;
#pragma unroll
    for (int e = 0; e < PT; ++e) {
        float Ae, Be;
        elemAB(t0 + e, Ae, Be);
        Bp = __builtin_fmaf(Ae, Bp, Be);
        Ap = Ae * Ap;
    }
    sA[tid] = Ap; sB[tid] = Bp;
    __syncthreads();

    // Hillis-Steele inclusive scan of thread aggregates
    for (int off = 1; off < BLOCK; off <<= 1) {
        float a1 = 1.0f, b1 = 0.0f;
        float a2 = sA[tid], b2 = sB[tid];
        bool has = (tid >= off);
        if (has) { a1 = sA[tid - off]; b1 = sB[tid - off]; }
        __syncthreads();
        if (has) { sA[tid] = a2 * a1; sB[tid] = __builtin_fmaf(a2, b1, b2); }
        __syncthreads();
    }

    if (!FINAL) {
        if (tid == BLOCK - 1) {
            agg[(row * nchunks + chk) * 2 + 0] = sA[tid];
            agg[(row * nchunks + chk) * 2 + 1] = sB[tid];
        }
        return;
    }

    // Carry-in y at chunk start; exclusive prefix for this thread
    float cin = carry[row * nchunks + chk];
    float Ax = 1.0f, Bx = 0.0f;
    if (tid > 0) { Ax = sA[tid - 1]; Bx = sB[tid - 1]; }
    float y = __builtin_fmaf(Ax, cin, Bx);

    float sumsq = 0.0f, cnt = 0.0f;
#pragma unroll
    for (int e = 0; e < PT; ++e) {
        float Ae, Be;
        elemAB(t0 + e, Ae, Be);
        y = __builtin_fmaf(Ae, y, Be);
        float Sp = sp_from_c(y);
        int g = rowbase + start + t0 + e;
        float m = (float)mask[g];
        float d = Sp - Sref[g];
        sumsq = __builtin_fmaf(m * d, d, sumsq);
        cnt  += m;
    }
    __syncthreads();
    sA[tid] = sumsq; sB[tid] = cnt;
    __syncthreads();
    for (int off = BLOCK / 2; off > 0; off >>= 1) {
        if (tid < off) { sA[tid] += sA[tid + off]; sB[tid] += sB[tid + off]; }
        __syncthreads();
    }
    if (tid == 0) {
        partial[blockIdx.x * 2 + 0] = sA[0];
        partial[blockIdx.x * 2 + 1] = sB[0];
    }
}

// One wave32 per row scans the chunk aggregates -> carry-in y per chunk.
__global__ void carry_scan_kernel(const float* __restrict__ agg,
                                  float* __restrict__ carry, int nchunks) {
    __shared__ float cA[256];
    __shared__ float cB[256];
    const int row = blockIdx.x;
    const int k   = threadIdx.x;
    cA[k] = agg[(row * nchunks + k) * 2 + 0];
    cB[k] = agg[(row * nchunks + k) * 2 + 1];
    __syncthreads();
    for (int off = 1; off < nchunks; off <<= 1) {
        float a1 = 1.0f, b1 = 0.0f;
        float a2 = cA[k], b2 = cB[k];
        bool has = (k >= off);
        if (has) { a1 = cA[k - off]; b1 = cB[k - off]; }
        __syncthreads();
        if (has) { cA[k] = a2 * a1; cB[k] = __builtin_fmaf(a2, b1, b2); }
        __syncthreads();
    }
    // carry for chunk k = y at end of chunk k-1 = inclusive B at k-1
    carry[row * nchunks + k] = (k == 0) ? 0.0f : cB[k - 1];
}

__global__ __launch_bounds__(256)
void finalize_kernel(const float* __restrict__ pd, const float* __restrict__ pu,
                     int nblk, const float* __restrict__ params, int nparams,
                     float* __restrict__ out) {
    __shared__ float r[5][256];
    const int tid = threadIdx.x;
    float sd = 0.f, cd = 0.f, su = 0.f, cu = 0.f, pr = 0.f;
    for (int i = tid; i < nblk; i += 256) {
        sd += pd[2 * i]; cd += pd[2 * i + 1];
        su += pu[2 * i]; cu += pu[2 * i + 1];
    }
    for (int i = tid; i < nparams; i += 256) { float p = params[i]; pr += p * p; }
    r[0][tid] = sd; r[1][tid] = cd; r[2][tid] = su; r[3][tid] = cu; r[4][tid] = pr;
    __syncthreads();
    for (int off = 128; off > 0; off >>= 1) {
        if (tid < off)
            for (int j = 0; j < 5; ++j) r[j][tid] += r[j][tid + off];
        __syncthreads();
    }
    if (tid == 0) {
        out[0] = r[0][0] / fmaxf(r[1][0], 1.0f)
               + r[2][0] / fmaxf(r[3][0], 1.0f)
               + 0.001f * (r[4][0] / (float)nparams);
    }
}

extern "C" void kernel_launch(void* const* d_in, const int* in_sizes, int n_in,
                              void* d_out, int out_size, void* d_ws, size_t ws_size,
                              hipStream_t stream) {
    // setup_inputs order: params, T_down, C_down, P_down, T_up, C_up, P_up,
    //                     S_ctd_down, S_ctd_up, mask_down, mask_up
    const float* params = (const float*)d_in[0];
    const float* Td  = (const float*)d_in[1];
    const float* Cd  = (const float*)d_in[2];
    const float* Tu  = (const float*)d_in[4];
    const float* Cu  = (const float*)d_in[5];
    const float* Srd = (const float*)d_in[7];
    const float* Sru = (const float*)d_in[8];
    const unsigned char* Md = (const unsigned char*)d_in[9];
    const unsigned char* Mu = (const unsigned char*)d_in[10];
    float* out = (float*)d_out;

    const int B      = 32;
    const int seqLen = in_sizes[1] / B;       // 131072
    const int nch    = seqLen / CHUNK;        // 32 chunks per row
    const int nblk   = B * nch;               // 1024 blocks per direction

    float* ws   = (float*)d_ws;
    float* aggD = ws;                 // [nblk][2]
    float* aggU = aggD + 2 * nblk;    // [nblk][2]
    float* carD = aggU + 2 * nblk;    // [nblk]
    float* carU = carD + nblk;        // [nblk]
    float* parD = carU + nblk;        // [nblk][2]
    float* parU = parD + 2 * nblk;    // [nblk][2]

    garau_scan_kernel<false><<<nblk, BLOCK, 0, stream>>>(
        Td, Cd, nullptr, nullptr, params, 0, seqLen, nch, aggD, nullptr, nullptr);
    garau_scan_kernel<false><<<nblk, BLOCK, 0, stream>>>(
        Tu, Cu, nullptr, nullptr, params, 4, seqLen, nch, aggU, nullptr, nullptr);
    carry_scan_kernel<<<B, nch, 0, stream>>>(aggD, carD, nch);
    carry_scan_kernel<<<B, nch, 0, stream>>>(aggU, carU, nch);
    garau_scan_kernel<true><<<nblk, BLOCK, 0, stream>>>(
        Td, Cd, Srd, Md, params, 0, seqLen, nch, nullptr, carD, parD);
    garau_scan_kernel<true><<<nblk, BLOCK, 0, stream>>>(
        Tu, Cu, Sru, Mu, params, 4, seqLen, nch, nullptr, carU, parU);
    finalize_kernel<<<1, 256, 0, stream>>>(parD, parU, nblk, params, in_sizes[0], out);
}